// IFIVAModule_69793218560271
// MI455X (gfx1250) — compile-verified
//
#include <hip/hip_runtime.h>
#include <hip/hip_bf16.h>
#include <math.h>

// Problem dims (from setup_inputs): B=8, K=512, d=8, N=2048
#define BDIM 8
#define KDIM 512
#define DDIM 8
#define NDIM 2048
#define BK   (BDIM * KDIM)
#define EPSF 1e-8f
#define NUM_ITERATIONS 5

// LDS tiling for the update pass
#define TILE 256
#define XROW (TILE * 2 + 4)   // padded row stride (floats): rows hit distinct banks

typedef __attribute__((ext_vector_type(2))) float v2f;
typedef __attribute__((ext_vector_type(8))) float v8f;

#if defined(__HIP_DEVICE_COMPILE__) && __has_builtin(__builtin_amdgcn_wmma_f32_16x16x4_f32)
#define HAVE_WMMA_F32X4 1
#else
#define HAVE_WMMA_F32X4 0
#endif

// ---------------------------------------------------------------------------
// Init: copy w_initial and a_initial into d_out (live w/a storage).
// d_out layout: [ w : BK*D*2 floats | a : BK*D*2 floats ]
// ---------------------------------------------------------------------------
__global__ void iva_init_kernel(const float* __restrict__ w0,
                                const float* __restrict__ a0,
                                float* __restrict__ out, int n) {
  int i = blockIdx.x * blockDim.x + threadIdx.x;
  if (i < n) {
    out[i]     = w0[i];
    out[n + i] = a0[i];
  }
}

__global__ void iva_zero_kernel(float* __restrict__ p, int n) {
  int i = blockIdx.x * blockDim.x + threadIdx.x;
  if (i < n) p[i] = 0.0f;
}

// ---------------------------------------------------------------------------
// Pass A: per (b,k) block. Computes sigma = sqrt(max(re(w^H Cx w), eps)),
// streams x once (coalesced) accumulating sum_k |soin|^2 into sum_sp2[b, n].
// ---------------------------------------------------------------------------
__global__ void iva_sigma_sp2_kernel(const float* __restrict__ x,
                                     const float* __restrict__ w,
                                     const float* __restrict__ Cx,
                                     float* __restrict__ sigma,
                                     float* __restrict__ sum_sp2) {
  __shared__ float wre[DDIM], wim[DDIM];
  __shared__ float sSig;
  const int bk = blockIdx.x;
  const int b  = bk / KDIM;
  const int t  = threadIdx.x;

  if (t < DDIM) {
    wre[t] = w[(bk * DDIM + t) * 2];
    wim[t] = w[(bk * DDIM + t) * 2 + 1];
  }
  __syncthreads();

  if (t == 0) {
    // sigmax2 = re( w^H (Cx @ w) ), d = 8 so do it serially.
    float s2 = 0.0f;
    const float* Cb = Cx + (size_t)bk * DDIM * DDIM * 2;
    for (int r = 0; r < DDIM; ++r) {
      float cr = 0.0f, ci = 0.0f;
      const float* row = Cb + (size_t)r * DDIM * 2;
      for (int c2 = 0; c2 < DDIM; ++c2) {
        float ar = row[c2 * 2], ai = row[c2 * 2 + 1];
        cr += ar * wre[c2] - ai * wim[c2];
        ci += ar * wim[c2] + ai * wre[c2];
      }
      s2 += wre[r] * cr + wim[r] * ci;   // re( conj(w[r]) * (cr + i ci) )
    }
    float sg = sqrtf(fmaxf(s2, EPSF));
    sSig = sg;
    sigma[bk] = sg;
  }
  __syncthreads();

  const float sg  = sSig;
  const float inv = sg / (sg * sg + EPSF);  // rdiv(z, sigma) scale (sigma real)
  const float* xb = x + (size_t)bk * DDIM * NDIM * 2;

  for (int n = t; n < NDIM; n += blockDim.x) {
    __builtin_prefetch(&xb[((size_t)n + blockDim.x) * 2], 0, 1);  // global_prefetch_b8
    float sr = 0.0f, si = 0.0f;
    for (int ch = 0; ch < DDIM; ++ch) {
      const float2 xv = *(const float2*)&xb[((size_t)ch * NDIM + n) * 2];
      sr += wre[ch] * xv.x + wim[ch] * xv.y;   // conj(w)*x
      si += wre[ch] * xv.y - wim[ch] * xv.x;
    }
    float pr = sr * inv, pi = si * inv;
    atomicAdd(&sum_sp2[b * NDIM + n], pr * pr + pi * pi);
  }
}

// 8x8 complex matvec: (yr,yi) = M @ (vr,vi); M interleaved row-major.
__device__ __forceinline__ void cmatvec8(const float* __restrict__ M,
                                         const float* vr, const float* vi,
                                         float* yr, float* yi) {
  for (int r = 0; r < DDIM; ++r) {
    float cr = 0.0f, ci = 0.0f;
    const float* row = M + (size_t)r * DDIM * 2;
    for (int c2 = 0; c2 < DDIM; ++c2) {
      float ar = row[c2 * 2], ai = row[c2 * 2 + 1];
      cr += ar * vr[c2] - ai * vi[c2];
      ci += ar * vi[c2] + ai * vr[c2];
    }
    yr[r] = cr; yi[r] = ci;
  }
}

// ---------------------------------------------------------------------------
// Pass C: per (b,k) block, LDS-tiled.
//   stage : coalesced copy of an 8ch x 256-frame x tile into LDS (x read 1x)
//   phase1: one thread per frame computes psi -> LDS, accumulates rho exactly
//   phase2: WMMA f32 16x16x4 accumulates xpsi_num = x @ psi^T
//           (real 16 x 2N GEMV, rows = [re(ch0..7); im(ch0..7)]) from LDS
//   tail  : thread 0 performs the d=8 complex update algebra, writes w, a.
// ---------------------------------------------------------------------------
__global__ void iva_update_kernel(const float* __restrict__ x,
                                  float* __restrict__ w,
                                  float* __restrict__ a,
                                  const float* __restrict__ Cx,
                                  const float* __restrict__ Cwinv,
                                  const float* __restrict__ sigma,
                                  const float* __restrict__ sum_sp2) {
  __shared__ float wre[DDIM], wim[DDIM];
  __shared__ float xt[DDIM * XROW];     // padded x tile (16.5 KB)
  __shared__ float psit[TILE * 2];      // psi tile, interleaved re/im (2 KB)
  __shared__ float sXre[DDIM], sXim[DDIM];
  __shared__ float sRho;
  const int bk = blockIdx.x;
  const int b  = bk / KDIM;
  const int t  = threadIdx.x;
  const int L  = t & 31;        // lane in wave32
  const int wv = t >> 5;        // wave id (8 waves of 32)

  if (t < DDIM) {
    wre[t] = w[(bk * DDIM + t) * 2];
    wim[t] = w[(bk * DDIM + t) * 2 + 1];
    sXre[t] = 0.0f; sXim[t] = 0.0f;
  }
  if (t == 0) sRho = 0.0f;
  __syncthreads();

  const float sg  = sigma[bk];
  const float inv = sg / (sg * sg + EPSF);
  const float* xb  = x + (size_t)bk * DDIM * NDIM * 2;
  const float* ssb = sum_sp2 + b * NDIM;

  const int  m      = L & 15;   // A-matrix row this lane feeds (16x4 striping)
  const int  ch     = m & 7;
  const bool isReal = (m < 8);

  float accRho = 0.0f;
  v8f c = {};                   // WMMA f32 16x16 accumulator (8 VGPRs)
#if !HAVE_WMMA_F32X4
  float fallbackAcc = 0.0f;
#endif

  for (int tile = 0; tile < NDIM / TILE; ++tile) {
    const int base = tile * TILE;

    // --- stage: coalesced copy of x tile into LDS (bank-conflict padded) ---
    for (int idx = t; idx < DDIM * TILE; idx += 256) {
      const int chs = idx >> 8;           // TILE == 256
      const int off = idx & (TILE - 1);
      __builtin_prefetch(&xb[((size_t)chs * NDIM + base + off + TILE) * 2], 0, 1);
      const float2 v = *(const float2*)&xb[((size_t)chs * NDIM + base + off) * 2];
      xt[chs * XROW + off * 2]     = v.x;
      xt[chs * XROW + off * 2 + 1] = v.y;
    }
    __syncthreads();

    // --- phase 1: one frame per thread: soin -> psi, exact rho count ---
    {
      const int nl = t;                    // 0..255
      float sr = 0.0f, si = 0.0f;
      for (int c2 = 0; c2 < DDIM; ++c2) {
        const float xr = xt[c2 * XROW + nl * 2];
        const float xi = xt[c2 * XROW + nl * 2 + 1];
        sr += wre[c2] * xr + wim[c2] * xi;
        si += wre[c2] * xi - wim[c2] * xr;
      }
      const float snr = sr * inv, sni = si * inv;        // soin
      const float aux = 1.0f / (1.0f + ssb[base + nl] + EPSF);
      const float pr  =  snr * aux;                      // psi = conj(soin)*aux
      const float pi  = -sni * aux;
      psit[nl * 2]     = pr;
      psit[nl * 2 + 1] = pi;
      accRho += aux - (pr * pr + pi * pi);               // psihpsi (real)
    }
    __syncthreads();

    // --- phase 2: WMMA accumulation of xpsi_num over this tile ---
    // Per step: lanes 0..15 -> frame nl0 (A cols 0,1), lanes 16..31 -> nl1.
    for (int s = 0; s < TILE / 16; ++s) {
      const int nl  = s * 16 + wv * 2 + (L >> 4);
      const float mxr = xt[ch * XROW + nl * 2];
      const float mxi = xt[ch * XROW + nl * 2 + 1];
      const float pr  = psit[nl * 2];
      const float pi  = psit[nl * 2 + 1];
      // A row m: real rows (m<8) = [xr, -xi], imag rows = [xi, xr]
      const float a0 = isReal ?  mxr : mxi;
      const float a1 = isReal ? -mxi : mxr;
#if HAVE_WMMA_F32X4
      v2f af; af[0] = a0; af[1] = a1;
      v2f bf; bf[0] = pr; bf[1] = pi;     // B rows replicated over 16 columns
      c = __builtin_amdgcn_wmma_f32_16x16x4_f32(
          /*neg_a=*/false, af, /*neg_b=*/false, bf,
          /*c_mod=*/(short)0, c, /*reuse_a=*/false, /*reuse_b=*/false);
#else
      fallbackAcc += a0 * pr + a1 * pi;
#endif
    }
    __syncthreads();   // protect xt/psit before next tile's staging
  }

  // --- extract xpsi_num partials ---
#if HAVE_WMMA_F32X4
  // D layout: VGPR r, lanes 0-15 -> row r (real), lanes 16-31 -> row r+8
  // (imag); all 16 columns identical, so lanes 0 and 16 suffice.
  if (L == 0)  for (int r = 0; r < DDIM; ++r) atomicAdd(&sXre[r], c[r]);
  if (L == 16) for (int r = 0; r < DDIM; ++r) atomicAdd(&sXim[r], c[r]);
#else
  if (isReal) atomicAdd(&sXre[ch], fallbackAcc);
  else        atomicAdd(&sXim[ch], fallbackAcc);
#endif
  atomicAdd(&sRho, accRho);          // ds_add_f32
  __syncthreads();

  if (t == 0) {
    const float rho = sRho / (float)NDIM;   // exact mean over frames

    // xpsi = rdiv(rdiv(xpsi_num, sigma), N)
    const float Nf = (float)NDIM;
    const float scale = (sg / (sg * sg + EPSF)) * (Nf / (Nf * Nf + EPSF));
    float xpr[DDIM], xpi[DDIM];
    for (int r = 0; r < DDIM; ++r) { xpr[r] = sXre[r] * scale; xpi[r] = sXim[r] * scale; }

    // nu = w^H xpsi
    float nur = 0.0f, nui = 0.0f;
    for (int r = 0; r < DDIM; ++r) {
      nur += wre[r] * xpr[r] + wim[r] * xpi[r];
      nui += wre[r] * xpi[r] - wim[r] * xpr[r];
    }
    const float dnu = nur * nur + nui * nui + EPSF;

    // grad = a - rdiv(xpsi, nu)
    float ar_[DDIM], ai_[DDIM], gr[DDIM], gi[DDIM];
    for (int r = 0; r < DDIM; ++r) {
      ar_[r] = a[(bk * DDIM + r) * 2];
      ai_[r] = a[(bk * DDIM + r) * 2 + 1];
      float tr = (xpr[r] * nur + xpi[r] * nui) / dnu;
      float ti = (xpi[r] * nur - xpr[r] * nui) / dnu;
      gr[r] = ar_[r] - tr;
      gi[r] = ai_[r] - ti;
    }

    // denom_a = 1 - rdiv(rho, nu)  (rho real)
    const float der = 1.0f - rho * nur / dnu;
    const float dei =         rho * nui / dnu;
    const float dd  = der * der + dei * dei + EPSF;

    // a = a - rdiv(grad, denom_a)
    for (int r = 0; r < DDIM; ++r) {
      float tr = (gr[r] * der + gi[r] * dei) / dd;
      float ti = (gi[r] * der - gr[r] * dei) / dd;
      ar_[r] -= tr;
      ai_[r] -= ti;
    }

    // w = Cwinv @ a
    float wr2[DDIM], wi2[DDIM];
    cmatvec8(Cwinv + (size_t)bk * DDIM * DDIM * 2, ar_, ai_, wr2, wi2);

    // sigmaw2 = re(rdiv(1, a^H w)) ; w *= sigmaw2
    float s2r = 0.0f, s2i = 0.0f;
    for (int r = 0; r < DDIM; ++r) {
      s2r += ar_[r] * wr2[r] + ai_[r] * wi2[r];
      s2i += ar_[r] * wi2[r] - ai_[r] * wr2[r];
    }
    const float sigw2 = s2r / (s2r * s2r + s2i * s2i + EPSF);
    for (int r = 0; r < DDIM; ++r) { wr2[r] *= sigw2; wi2[r] *= sigw2; }

    // a = Cx @ w ; a = rdiv(a, a^H w)
    float anr[DDIM], ani[DDIM];
    cmatvec8(Cx + (size_t)bk * DDIM * DDIM * 2, wr2, wi2, anr, ani);
    float t2r = 0.0f, t2i = 0.0f;
    for (int r = 0; r < DDIM; ++r) {
      t2r += anr[r] * wr2[r] + ani[r] * wi2[r];
      t2i += anr[r] * wi2[r] - ani[r] * wr2[r];
    }
    const float td = t2r * t2r + t2i * t2i + EPSF;
    for (int r = 0; r < DDIM; ++r) {
      float fr = (anr[r] * t2r + ani[r] * t2i) / td;
      float fi = (ani[r] * t2r - anr[r] * t2i) / td;
      a[(bk * DDIM + r) * 2]     = fr;
      a[(bk * DDIM + r) * 2 + 1] = fi;
      w[(bk * DDIM + r) * 2]     = wr2[r];
      w[(bk * DDIM + r) * 2 + 1] = wi2[r];
    }
  }
}

// ---------------------------------------------------------------------------
extern "C" void kernel_launch(void* const* d_in, const int* in_sizes, int n_in,
                              void* d_out, int out_size, void* d_ws, size_t ws_size,
                              hipStream_t stream) {
  (void)in_sizes; (void)n_in; (void)out_size; (void)ws_size;
  const float* x     = (const float*)d_in[0];  // (B,K,d,N) complex64 interleaved
  const float* w0    = (const float*)d_in[1];  // (B,K,d,1) complex64
  const float* a0    = (const float*)d_in[2];  // (B,K,d,1) complex64
  const float* Cx    = (const float*)d_in[3];  // (B,K,d,d) complex64
  const float* Cwinv = (const float*)d_in[4];  // (B,K,d,d) complex64

  float* out = (float*)d_out;
  float* w = out;                          // BK*D*2 floats
  float* a = out + BK * DDIM * 2;          // BK*D*2 floats

  float* sigma   = (float*)d_ws;           // BK floats
  float* sum_sp2 = sigma + BK;             // B*N floats

  const int wa_n = BK * DDIM * 2;
  iva_init_kernel<<<(wa_n + 255) / 256, 256, 0, stream>>>(w0, a0, out, wa_n);

  for (int it = 0; it < NUM_ITERATIONS; ++it) {
    iva_zero_kernel<<<(BDIM * NDIM + 255) / 256, 256, 0, stream>>>(sum_sp2, BDIM * NDIM);
    iva_sigma_sp2_kernel<<<BK, 256, 0, stream>>>(x, w, Cx, sigma, sum_sp2);
    iva_update_kernel<<<BK, 256, 0, stream>>>(x, w, a, Cx, Cwinv, sigma, sum_sp2);
  }
}